// GCNEncoder_29274497089997
// MI455X (gfx1250) — compile-verified
//
#include <hip/hip_runtime.h>

typedef __attribute__((ext_vector_type(2))) float v2f;
typedef __attribute__((ext_vector_type(8))) float v8f;

#define DF 128          // feature dim (fixed by reference)
#define SXS 132         // padded LDS stride for X tile (kills bank conflicts)

static __device__ __forceinline__ void atomAddF(float* p, float v) {
    // Hardware global_atomic_add_f32 (no CAS loop)
    unsafeAtomicAdd(p, v);
}

// ---------------------------------------------------------------- utilities
__global__ void k_zero(float* __restrict__ p, int n) {
    int i = blockIdx.x * blockDim.x + threadIdx.x;
    if (i < n) p[i] = 0.0f;
}

__global__ void k_deg(const int* __restrict__ src, const int* __restrict__ dst,
                      float* __restrict__ degS, float* __restrict__ degD, int E) {
    int i = blockIdx.x * blockDim.x + threadIdx.x;
    if (i < E) {
        atomAddF(&degS[src[i]], 1.0f);
        atomAddF(&degD[dst[i]], 1.0f);
    }
}

__global__ void k_rinv(float* __restrict__ p, int n) {
    int i = blockIdx.x * blockDim.x + threadIdx.x;
    if (i < n) {
        float d = p[i];
        d = d < 1.0f ? 1.0f : d;      // clip(deg, 1.0)
        p[i] = rsqrtf(d);
    }
}

__global__ void k_bias_init(float* __restrict__ h, const float* __restrict__ b, int total) {
    int i = blockIdx.x * blockDim.x + threadIdx.x;
    if (i < total) h[i] = b[i & (DF - 1)];
}

__global__ void k_relu(float* __restrict__ p, int n) {
    int i = blockIdx.x * blockDim.x + threadIdx.x;
    if (i < n) {
        float v = p[i];
        p[i] = v > 0.0f ? v : 0.0f;
    }
}

// ---------------------------------------------------------------- GEMM
// Y[nrows,128] = X[nrows,128] @ W[128,128]  (row-major, fp32, WMMA 16x16x4)
// Block: 256 threads = 8 wave32. Block covers 16 rows x 128 cols; each wave
// owns one 16x16 tile (wave w -> cols [16w,16w+16)).
__global__ __launch_bounds__(256)
void k_gemm128(const float* __restrict__ X, const float* __restrict__ W,
               float* __restrict__ Y, int nrows) {
    __shared__ float sW[DF * DF];        // 64 KB
    __shared__ float sX[16 * SXS];       // padded 16x128 band

    const int tid  = threadIdx.x;
    const int row0 = blockIdx.x * 16;

    // Stage W (16384 floats) via b128, fully coalesced.
    {
        const float4* Wv  = (const float4*)W;
        float4*       sWv = (float4*)sW;
#pragma unroll
        for (int i = 0; i < 16; ++i) sWv[tid + 256 * i] = Wv[tid + 256 * i];
    }
    // Stage X band (2048 floats) into padded LDS; zero-fill past nrows.
    {
        const float4* Xv = (const float4*)(X + (long)row0 * DF);
#pragma unroll
        for (int i = 0; i < 2; ++i) {
            int idx = tid + 256 * i;     // float4 index in 16x32 grid
            int r = idx >> 5;            // row 0..15
            int c = idx & 31;            // float4 col 0..31
            float4 v = make_float4(0.f, 0.f, 0.f, 0.f);
            if (row0 + r < nrows) v = Xv[idx];
            *(float4*)&sX[r * SXS + c * 4] = v;   // (r*528 + c*16) bytes: 16B aligned
        }
    }
    __syncthreads();

    const int lane    = tid & 31;
    const int wave    = tid >> 5;
    const int half    = lane >> 4;       // 0: lanes 0-15, 1: lanes 16-31
    const int l16     = lane & 15;
    const int colbase = wave * 16;

    v8f c = {0.f, 0.f, 0.f, 0.f, 0.f, 0.f, 0.f, 0.f};

#pragma unroll
    for (int k0 = 0; k0 < DF; k0 += 4) {
        const int ka = k0 + 2 * half;
        // A (16x4 f32): lanes 0-15 hold M=l16, K={k0,k0+1}; lanes 16-31 K={k0+2,k0+3}
        v2f a = *(const v2f*)&sX[l16 * SXS + ka];
        // B (4x16 f32): mirrored layout, row K striped across lanes within a VGPR
        v2f b;
        b.x = sW[ka * DF + colbase + l16];
        b.y = sW[(ka + 1) * DF + colbase + l16];
        c = __builtin_amdgcn_wmma_f32_16x16x4_f32(
                /*neg_a=*/false, a, /*neg_b=*/false, b,
                /*c_mod=*/(short)0, c, /*reuse_a=*/false, /*reuse_b=*/false);
    }

    // C/D layout: lanes 0-15 -> M = r; lanes 16-31 -> M = r + 8 (VGPR r = c[r])
    float* Yp = Y + (long)row0 * DF + colbase + l16;
#pragma unroll
    for (int r = 0; r < 8; ++r) {
        int m = r + 8 * half;
        if (row0 + m < nrows) Yp[(long)m * DF] = c[r];
    }
}

// ---------------------------------------------------------------- scatter
// One wave per edge: H[dst] += (XW)[src] * rsqrt(deg_s[src]) * rsqrt(deg_d[dst])
__global__ __launch_bounds__(256)
void k_scatter(const float* __restrict__ Ysrc,
               const int* __restrict__ src, const int* __restrict__ dst,
               const float* __restrict__ rinvS, const float* __restrict__ rinvD,
               float* __restrict__ H, int E) {
    int gid  = blockIdx.x * blockDim.x + threadIdx.x;
    int e    = gid >> 5;
    int lane = gid & 31;
    if (e >= E) return;
    int s = src[e], d = dst[e];
    float norm = rinvS[s] * rinvD[d];
    float4 v = ((const float4*)(Ysrc + (long)s * DF))[lane];   // 512B/wave, coalesced
    float* hp = H + (long)d * DF + lane * 4;
    atomAddF(hp + 0, v.x * norm);
    atomAddF(hp + 1, v.y * norm);
    atomAddF(hp + 2, v.z * norm);
    atomAddF(hp + 3, v.w * norm);
}

// ---------------------------------------------------------------- launch
extern "C" void kernel_launch(void* const* d_in, const int* in_sizes, int n_in,
                              void* d_out, int out_size, void* d_ws, size_t ws_size,
                              hipStream_t stream) {
    const float* x_lit = (const float*)d_in[0];
    const float* x_cl  = (const float*)d_in[1];
    const float* W0_lc = (const float*)d_in[2];
    const float* b0_lc = (const float*)d_in[3];
    const float* W0_cl = (const float*)d_in[4];
    const float* b0_cl = (const float*)d_in[5];
    const float* W1_lc = (const float*)d_in[6];
    const float* b1_lc = (const float*)d_in[7];
    const float* W1_cl = (const float*)d_in[8];
    const float* b1_cl = (const float*)d_in[9];
    const int* e_s_lc = (const int*)d_in[10];
    const int* e_d_lc = (const int*)d_in[11];
    const int* e_s_cl = (const int*)d_in[12];
    const int* e_d_cl = (const int*)d_in[13];

    const int N_LIT = in_sizes[0] / DF;   // 50000
    const int N_CL  = in_sizes[1] / DF;   // 200000
    const int E     = in_sizes[10];       // 600000

    // -------- workspace layout (floats) --------
    float* ws = (float*)d_ws;
    float* rinv_lit_s = ws;                          // [N_LIT]  deg of edge_src_lc
    float* rinv_cl_d  = rinv_lit_s + N_LIT;          // [N_CL]   deg of edge_dst_lc
    float* rinv_cl_s  = rinv_cl_d + N_CL;            // [N_CL]   deg of edge_src_cl
    float* rinv_lit_d = rinv_cl_s + N_CL;            // [N_LIT]  deg of edge_dst_cl
    float* Y_lit = rinv_lit_d + N_LIT;               // [N_LIT*128]
    float* Y_cl  = Y_lit + (long)N_LIT * DF;         // [N_CL*128]
    float* H_lit = Y_cl + (long)N_CL * DF;           // [N_LIT*128] layer-0 out
    float* H_cl  = H_lit + (long)N_LIT * DF;         // [N_CL*128]  layer-0 out

    float* out_lit = (float*)d_out;                  // tuple order: (h_lit, h_cl)
    float* out_cl  = out_lit + (long)N_LIT * DF;

    const int B = 256;
    const int degTot = 2 * (N_LIT + N_CL);
    const int nLitF  = N_LIT * DF;
    const int nClF   = N_CL * DF;
    const int scGrid = (E * 32 + B - 1) / B;

    // -------- degrees (shared by both layers) --------
    k_zero<<<(degTot + B - 1) / B, B, 0, stream>>>(ws, degTot);
    k_deg <<<(E + B - 1) / B, B, 0, stream>>>(e_s_lc, e_d_lc, rinv_lit_s, rinv_cl_d, E);
    k_deg <<<(E + B - 1) / B, B, 0, stream>>>(e_s_cl, e_d_cl, rinv_cl_s, rinv_lit_d, E);
    k_rinv<<<(degTot + B - 1) / B, B, 0, stream>>>(ws, degTot);

    // -------- layer 0 --------
    k_gemm128<<<(N_LIT + 15) / 16, B, 0, stream>>>(x_lit, W0_lc, Y_lit, N_LIT);
    k_gemm128<<<(N_CL + 15) / 16, B, 0, stream>>>(x_cl, W0_cl, Y_cl, N_CL);
    k_bias_init<<<(nClF + B - 1) / B, B, 0, stream>>>(H_cl, b0_lc, nClF);
    k_bias_init<<<(nLitF + B - 1) / B, B, 0, stream>>>(H_lit, b0_cl, nLitF);
    k_scatter<<<scGrid, B, 0, stream>>>(Y_lit, e_s_lc, e_d_lc, rinv_lit_s, rinv_cl_d, H_cl, E);
    k_scatter<<<scGrid, B, 0, stream>>>(Y_cl, e_s_cl, e_d_cl, rinv_cl_s, rinv_lit_d, H_lit, E);
    k_relu<<<(nLitF + nClF + B - 1) / B, B, 0, stream>>>(H_lit, nLitF + nClF); // H_lit,H_cl contiguous

    // -------- layer 1 (scatter straight into d_out) --------
    k_gemm128<<<(N_LIT + 15) / 16, B, 0, stream>>>(H_lit, W1_lc, Y_lit, N_LIT);
    k_gemm128<<<(N_CL + 15) / 16, B, 0, stream>>>(H_cl, W1_cl, Y_cl, N_CL);
    k_bias_init<<<(nClF + B - 1) / B, B, 0, stream>>>(out_cl, b1_lc, nClF);
    k_bias_init<<<(nLitF + B - 1) / B, B, 0, stream>>>(out_lit, b1_cl, nLitF);
    k_scatter<<<scGrid, B, 0, stream>>>(Y_lit, e_s_lc, e_d_lc, rinv_lit_s, rinv_cl_d, out_cl, E);
    k_scatter<<<scGrid, B, 0, stream>>>(Y_cl, e_s_cl, e_d_cl, rinv_cl_s, rinv_lit_d, out_lit, E);
    k_relu<<<(out_size + B - 1) / B, B, 0, stream>>>((float*)d_out, out_size);
}